// c_nn_50508815401523
// MI455X (gfx1250) — compile-verified
//
#include <hip/hip_runtime.h>
#include <math.h>

// ---------------------------------------------------------------------------
// MI455X (gfx1250) implementation.
// Bandwidth-bound workload (~220 MB of fp32 weights vs ~12 GFLOP), so all
// GEMMs use V_WMMA_F32_16X16X4_F32 (fp32 in/out: no precision loss, and no
// bandwidth to be saved by narrower types since weights are fp32 in HBM).
// ---------------------------------------------------------------------------

typedef float v2f __attribute__((ext_vector_type(2)));
typedef float v8f __attribute__((ext_vector_type(8)));

#define EPS 1e-5f

// ---------------------------------------------------------------------------
// conv3x3 (pad 1) + bias + relu + maxpool2x2, direct. in:(64,Cin,H,W)
// out:(64,Cout,H/2,W/2)
// ---------------------------------------------------------------------------
__global__ void conv3x3_relu_pool(const float* __restrict__ in,
                                  const float* __restrict__ w,
                                  const float* __restrict__ bias,
                                  float* __restrict__ out,
                                  int Cin, int Cout, int H, int Wd) {
    const int Hp = H >> 1, Wp = Wd >> 1;
    const int total = 64 * Cout * Hp * Wp;
    int idx = blockIdx.x * blockDim.x + threadIdx.x;
    if (idx >= total) return;
    int pw = idx % Wp; int t = idx / Wp;
    int ph = t % Hp;  t /= Hp;
    int co = t % Cout; int n = t / Cout;

    float mx = 0.0f;  // relu output >= 0, so pool max floor 0 is safe
    for (int dy = 0; dy < 2; ++dy) {
        for (int dx = 0; dx < 2; ++dx) {
            int oy = ph * 2 + dy, ox = pw * 2 + dx;
            float s = bias[co];
            for (int ci = 0; ci < Cin; ++ci) {
                const float* ip = in + ((long)(n * Cin + ci)) * H * Wd;
                const float* wp = w + ((long)(co * Cin + ci)) * 9;
                for (int ky = 0; ky < 3; ++ky) {
                    int y = oy + ky - 1;
                    if (y < 0 || y >= H) continue;
                    for (int kx = 0; kx < 3; ++kx) {
                        int x = ox + kx - 1;
                        if (x < 0 || x >= Wd) continue;
                        s += ip[y * Wd + x] * wp[ky * 3 + kx];
                    }
                }
            }
            s = fmaxf(s, 0.0f);
            mx = fmaxf(mx, s);
        }
    }
    out[idx] = mx;
}

// ---------------------------------------------------------------------------
// Per-channel BN statistics over (N, Hp, Wp). y:(64,C,HW). One block per ch.
// ---------------------------------------------------------------------------
__global__ void bn_stats(const float* __restrict__ y, float* __restrict__ mean,
                         float* __restrict__ rstd, int C, int HW) {
    const int c = blockIdx.x;
    const int cnt = 64 * HW;
    float s = 0.f, sq = 0.f;
    for (int i = threadIdx.x; i < cnt; i += blockDim.x) {
        int n = i / HW, j = i - n * HW;
        float v = y[((long)n * C + c) * HW + j];
        s += v; sq += v * v;
    }
    __shared__ float rs[256], rq[256];
    rs[threadIdx.x] = s; rq[threadIdx.x] = sq;
    __syncthreads();
    for (int off = 128; off > 0; off >>= 1) {
        if (threadIdx.x < off) {
            rs[threadIdx.x] += rs[threadIdx.x + off];
            rq[threadIdx.x] += rq[threadIdx.x + off];
        }
        __syncthreads();
    }
    if (threadIdx.x == 0) {
        float m = rs[0] / (float)cnt;
        float var = rq[0] / (float)cnt - m * m;   // population variance (ddof=0)
        mean[c] = m;
        rstd[c] = rsqrtf(var + EPS);
    }
}

__global__ void bn_apply(float* __restrict__ y, const float* __restrict__ mean,
                         const float* __restrict__ rstd,
                         const float* __restrict__ g, const float* __restrict__ be,
                         int C, int HW, int total) {
    int idx = blockIdx.x * blockDim.x + threadIdx.x;
    if (idx >= total) return;
    int c = (idx / HW) % C;
    y[idx] = (y[idx] - mean[c]) * rstd[c] * g[c] + be[c];
}

// ---------------------------------------------------------------------------
// Batched GEMM via V_WMMA_F32_16X16X4_F32.
//   C[batch] (64 x N) = relu?( A[batch>>aShift] (64 x K) * W[batch]^T + bias )
// A row-major 64xK, W row-major NxK. K must be a multiple of 64 (true here).
// Workgroup = 256 threads = 8 waves; wave tile 16x16; block tile 64(M)x32(N).
// grid.x = ceil(N/32), grid.y = batch count.
// ---------------------------------------------------------------------------
__global__ void __launch_bounds__(256)
gemm_wmma_f32(const float* __restrict__ A, const float* __restrict__ W,
              const float* __restrict__ bias, float* __restrict__ C,
              int N, int K,
              long strideA, long strideW, long strideBias, long strideC,
              int aShift, int doRelu) {
    __shared__ float As[64][68];   // 68-f32 pitch: 16B-aligned rows, staggered banks
    __shared__ float Bs[32][68];

    const int tid  = threadIdx.x;
    const int lane = tid & 31;            // wave32
    const int wave = tid >> 5;
    const int wm = (wave & 3) * 16;       // wave's M offset in block tile
    const int wn = (wave >> 2) * 16;      // wave's N offset in block tile
    const int nBlock = blockIdx.x * 32;
    const int batch  = blockIdx.y;

    const float* Ab = A + (long)(batch >> aShift) * strideA;
    const float* Wb = W + (long)batch * strideW;

    const int mi = wm + (lane & 15);      // A frag: lanes 0-15 m, k={0,1}; 16-31 k={2,3}
    const int ni = wn + (lane & 15);      // B frag: same striping on N
    const int ko = (lane >> 4) << 1;

    v8f acc = {};

    for (int k0 = 0; k0 < K; k0 += 64) {
        // Prefetch next W slab (memory-bound: keep HBM pipe full).
        if (k0 + 64 < K) {
            int pn = nBlock + (tid >> 3);
            if (pn < N)
                __builtin_prefetch(&Wb[(long)pn * K + k0 + 64 + ((tid & 7) << 3)], 0, 1);
        }
        // Stage A tile 64x64 (16 KB), 128-bit coalesced along K.
        #pragma unroll
        for (int i = 0; i < 4; ++i) {
            int lin = tid + i * 256;
            int r = lin >> 4, c4 = (lin & 15) << 2;
            float4 v = *reinterpret_cast<const float4*>(&Ab[(long)r * K + k0 + c4]);
            *reinterpret_cast<float4*>(&As[r][c4]) = v;
        }
        // Stage W tile 32x64 (8 KB), guarded on N (output layer N=1000).
        #pragma unroll
        for (int i = 0; i < 2; ++i) {
            int lin = tid + i * 256;
            int r = lin >> 4, c4 = (lin & 15) << 2;
            int n = nBlock + r;
            float4 v;
            if (n < N) {
                v = *reinterpret_cast<const float4*>(&Wb[(long)n * K + k0 + c4]);
            } else {
                v.x = v.y = v.z = v.w = 0.0f;
            }
            *reinterpret_cast<float4*>(&Bs[r][c4]) = v;
        }
        __syncthreads();

        #pragma unroll
        for (int kk = 0; kk < 64; kk += 4) {
            v2f a, b;
            a.x = As[mi][kk + ko];
            a.y = As[mi][kk + ko + 1];
            b.x = Bs[ni][kk + ko];
            b.y = Bs[ni][kk + ko + 1];
            acc = __builtin_amdgcn_wmma_f32_16x16x4_f32(
                /*neg_a=*/false, a, /*neg_b=*/false, b,
                /*c_mod=*/(short)0, acc, /*reuse_a=*/false, /*reuse_b=*/false);
        }
        __syncthreads();
    }

    // Epilogue: C/D layout -> VGPR r: lanes 0-15 hold M=wm+r, lanes 16-31 M=wm+r+8.
    const int nc = nBlock + wn + (lane & 15);
    if (nc < N) {
        float bv = bias ? bias[(long)batch * strideBias + nc] : 0.0f;
        #pragma unroll
        for (int r = 0; r < 8; ++r) {
            int m = wm + r + ((lane >> 4) << 3);
            float v = acc[r] + bv;
            if (doRelu) v = fmaxf(v, 0.0f);
            C[(long)batch * strideC + (long)m * N + nc] = v;
        }
    }
}

// ---------------------------------------------------------------------------
// Router 0: r0sel[b] = softmax(x_b @ rw0^T + rb0)[0], K=8192. One block per b.
// ---------------------------------------------------------------------------
__global__ void router0(const float* __restrict__ x, const float* __restrict__ rw,
                        const float* __restrict__ rb, float* __restrict__ r0sel) {
    const int b = blockIdx.x;
    float a0 = 0.f, a1 = 0.f;
    for (int i = threadIdx.x; i < 8192; i += 256) {
        float xv = x[(long)b * 8192 + i];
        a0 += xv * rw[i];
        a1 += xv * rw[8192 + i];
    }
    __shared__ float s0[256], s1[256];
    s0[threadIdx.x] = a0; s1[threadIdx.x] = a1;
    __syncthreads();
    for (int off = 128; off > 0; off >>= 1) {
        if (threadIdx.x < off) {
            s0[threadIdx.x] += s0[threadIdx.x + off];
            s1[threadIdx.x] += s1[threadIdx.x + off];
        }
        __syncthreads();
    }
    if (threadIdx.x == 0) {
        float z0 = s0[0] + rb[0], z1 = s1[0] + rb[1];
        float mx = fmaxf(z0, z1);
        float e0 = expf(z0 - mx), e1 = expf(z1 - mx);
        r0sel[b] = e0 / (e0 + e1);   // reference bug: children scaled by r0[:,0]
    }
}

// ---------------------------------------------------------------------------
// BN-feat over batch axis for h0:(2,64,2048), then scale by r0sel[b]. In place.
// One thread per feature f = k*2048 + o (4096 total).
// ---------------------------------------------------------------------------
__global__ void bnfeat0(float* __restrict__ h, const float* __restrict__ g,
                        const float* __restrict__ be, const float* __restrict__ r0sel) {
    int f = blockIdx.x * blockDim.x + threadIdx.x;
    if (f >= 4096) return;
    int k = f >> 11, o = f & 2047;
    long base = (long)k * 131072 + o;
    float s = 0.f, sq = 0.f;
    for (int b = 0; b < 64; ++b) {
        float v = h[base + (long)b * 2048];
        s += v; sq += v * v;
    }
    float m = s * (1.0f / 64.0f);
    float var = sq * (1.0f / 64.0f) - m * m;
    float rs = rsqrtf(var + EPS);
    float gg = g[f], bb = be[f];
    for (int b = 0; b < 64; ++b) {
        long i = base + (long)b * 2048;
        h[i] = ((h[i] - m) * rs * gg + bb) * r0sel[b];
    }
}

// ---------------------------------------------------------------------------
// Router 1: r1sel[n*64+b] = softmax(h0[n,b,:] @ rw1[n]^T + rb1[n])[1], K=2048.
// One block per (n,b).
// ---------------------------------------------------------------------------
__global__ void router1(const float* __restrict__ h0, const float* __restrict__ rw1,
                        const float* __restrict__ rb1, float* __restrict__ r1sel) {
    const int nb = blockIdx.x;
    const int n = nb >> 6, b = nb & 63;
    const float* hp = h0 + ((long)n * 64 + b) * 2048;
    const float* w0 = rw1 + (long)n * 4096;
    const float* w1 = w0 + 2048;
    float a0 = 0.f, a1 = 0.f;
    for (int i = threadIdx.x; i < 2048; i += 256) {
        float hv = hp[i];
        a0 += hv * w0[i];
        a1 += hv * w1[i];
    }
    __shared__ float s0[256], s1[256];
    s0[threadIdx.x] = a0; s1[threadIdx.x] = a1;
    __syncthreads();
    for (int off = 128; off > 0; off >>= 1) {
        if (threadIdx.x < off) {
            s0[threadIdx.x] += s0[threadIdx.x + off];
            s1[threadIdx.x] += s1[threadIdx.x + off];
        }
        __syncthreads();
    }
    if (threadIdx.x == 0) {
        float z0 = s0[0] + rb1[n * 2 + 0], z1 = s1[0] + rb1[n * 2 + 1];
        float mx = fmaxf(z0, z1);
        float e0 = expf(z0 - mx), e1 = expf(z1 - mx);
        r1sel[nb] = e1 / (e0 + e1);  // reference bug: scale by r1[n,:,1]
    }
}

// ---------------------------------------------------------------------------
// BN-feat for h1:(4,64,1024) (=(n,k,b,o)), scale by r1sel[n,b], and write
// transposed into leaves:(64,4096) with column nk*1024+o.
// ---------------------------------------------------------------------------
__global__ void bnfeat1(const float* __restrict__ h, const float* __restrict__ g,
                        const float* __restrict__ be, const float* __restrict__ r1sel,
                        float* __restrict__ leaves) {
    int f = blockIdx.x * blockDim.x + threadIdx.x;
    if (f >= 4096) return;
    int nk = f >> 10;
    int n = nk >> 1;
    int o = f & 1023;
    long base = (long)nk * 65536 + o;
    float s = 0.f, sq = 0.f;
    for (int b = 0; b < 64; ++b) {
        float v = h[base + (long)b * 1024];
        s += v; sq += v * v;
    }
    float m = s * (1.0f / 64.0f);
    float var = sq * (1.0f / 64.0f) - m * m;
    float rs = rsqrtf(var + EPS);
    float gg = g[f], bb = be[f];
    for (int b = 0; b < 64; ++b) {
        float v = ((h[base + (long)b * 1024] - m) * rs * gg + bb) * r1sel[n * 64 + b];
        leaves[(long)b * 4096 + f] = v;
    }
}

// ---------------------------------------------------------------------------
// Host launch sequence.
// ---------------------------------------------------------------------------
extern "C" void kernel_launch(void* const* d_in, const int* in_sizes, int n_in,
                              void* d_out, int out_size, void* d_ws, size_t ws_size,
                              hipStream_t stream) {
    const float* x      = (const float*)d_in[0];
    const float* cw0    = (const float*)d_in[1];
    const float* cb0    = (const float*)d_in[2];
    const float* g0     = (const float*)d_in[3];
    const float* be0    = (const float*)d_in[4];
    const float* cw1    = (const float*)d_in[5];
    const float* cb1    = (const float*)d_in[6];
    const float* g1     = (const float*)d_in[7];
    const float* be1    = (const float*)d_in[8];
    const float* cw2    = (const float*)d_in[9];
    const float* cb2    = (const float*)d_in[10];
    const float* g2     = (const float*)d_in[11];
    const float* be2    = (const float*)d_in[12];
    const float* tw0    = (const float*)d_in[13];
    const float* tbias0 = (const float*)d_in[14];
    const float* tg0    = (const float*)d_in[15];
    const float* tbe0   = (const float*)d_in[16];
    const float* rw0    = (const float*)d_in[17];
    const float* rb0    = (const float*)d_in[18];
    const float* tw1    = (const float*)d_in[19];
    const float* tbias1 = (const float*)d_in[20];
    const float* tg1    = (const float*)d_in[21];
    const float* tbe1   = (const float*)d_in[22];
    const float* rw1    = (const float*)d_in[23];
    const float* rb1    = (const float*)d_in[24];
    const float* fw0    = (const float*)d_in[25];
    const float* fb0    = (const float*)d_in[26];
    const float* fw1    = (const float*)d_in[27];
    const float* fb1    = (const float*)d_in[28];
    const float* ow     = (const float*)d_in[29];
    const float* ob     = (const float*)d_in[30];
    float* out = (float*)d_out;

    float* ws = (float*)d_ws;
    float* c0buf  = ws;                    // 64*32*32*32 = 2097152
    float* c1buf  = c0buf  + 2097152;      // 64*64*16*16 = 1048576
    float* c2buf  = c1buf  + 1048576;      // 64*128*8*8  = 524288  (= x_flat 64x8192)
    float* mean   = c2buf  + 524288;       // 128
    float* rstd   = mean   + 128;          // 128
    float* h0     = rstd   + 128;          // (2,64,2048) = 262144
    float* r0sel  = h0     + 262144;       // 64
    float* h1     = r0sel  + 64;           // (4,64,1024) = 262144
    float* r1sel  = h1     + 262144;       // 128
    float* leaves = r1sel  + 128;          // (64,4096)   = 262144
    float* f0     = leaves + 262144;       // (64,2048)   = 131072
    float* f1     = f0     + 131072;       // (64,1024)   = 65536
    (void)ws_size; (void)n_in; (void)in_sizes; (void)out_size;

    // --- conv block 0: (64,3,64,64) -> (64,32,32,32) ---
    conv3x3_relu_pool<<<8192, 256, 0, stream>>>(x, cw0, cb0, c0buf, 3, 32, 64, 64);
    bn_stats<<<32, 256, 0, stream>>>(c0buf, mean, rstd, 32, 1024);
    bn_apply<<<8192, 256, 0, stream>>>(c0buf, mean, rstd, g0, be0, 32, 1024, 2097152);

    // --- conv block 1: -> (64,64,16,16) ---
    conv3x3_relu_pool<<<4096, 256, 0, stream>>>(c0buf, cw1, cb1, c1buf, 32, 64, 32, 32);
    bn_stats<<<64, 256, 0, stream>>>(c1buf, mean, rstd, 64, 256);
    bn_apply<<<4096, 256, 0, stream>>>(c1buf, mean, rstd, g1, be1, 64, 256, 1048576);

    // --- conv block 2: -> (64,128,8,8) == x_flat (64,8192) ---
    conv3x3_relu_pool<<<2048, 256, 0, stream>>>(c1buf, cw2, cb2, c2buf, 64, 128, 16, 16);
    bn_stats<<<128, 256, 0, stream>>>(c2buf, mean, rstd, 128, 64);
    bn_apply<<<2048, 256, 0, stream>>>(c2buf, mean, rstd, g2, be2, 128, 64, 524288);

    // --- tree depth 0: h0 = relu(x_flat @ tw0^T + tbias0), (2,64,2048) ---
    // batch over k=0,1 (shared A via aShift=1); this is the 134 MB weight read.
    gemm_wmma_f32<<<dim3(64, 2), 256, 0, stream>>>(
        c2buf, tw0, tbias0, h0, 2048, 8192,
        /*strideA=*/0, /*strideW=*/(long)2048 * 8192, /*strideBias=*/2048,
        /*strideC=*/(long)64 * 2048, /*aShift=*/1, /*relu=*/1);

    router0<<<64, 256, 0, stream>>>(c2buf, rw0, rb0, r0sel);
    bnfeat0<<<16, 256, 0, stream>>>(h0, tg0, tbe0, r0sel);

    // --- tree depth 1: batch over (n,k); A = h0[n] via aShift=1 ---
    router1<<<128, 256, 0, stream>>>(h0, rw1, rb1, r1sel);
    gemm_wmma_f32<<<dim3(32, 4), 256, 0, stream>>>(
        h0, tw1, tbias1, h1, 1024, 2048,
        /*strideA=*/(long)64 * 2048, /*strideW=*/(long)1024 * 2048, /*strideBias=*/1024,
        /*strideC=*/(long)64 * 1024, /*aShift=*/1, /*relu=*/1);

    bnfeat1<<<16, 256, 0, stream>>>(h1, tg1, tbe1, r1sel, leaves);

    // --- FC stack ---
    gemm_wmma_f32<<<dim3(64, 1), 256, 0, stream>>>(
        leaves, fw0, fb0, f0, 2048, 4096, 0, 0, 0, 0, /*aShift=*/0, /*relu=*/1);
    gemm_wmma_f32<<<dim3(32, 1), 256, 0, stream>>>(
        f0, fw1, fb1, f1, 1024, 2048, 0, 0, 0, 0, /*aShift=*/0, /*relu=*/1);
    gemm_wmma_f32<<<dim3(32, 1), 256, 0, stream>>>(
        f1, ow, ob, out, 1000, 1024, 0, 0, 0, 0, /*aShift=*/0, /*relu=*/0);
}